// NormalizedCrossCorrelation_52243982188891
// MI455X (gfx1250) — compile-verified
//
#include <hip/hip_runtime.h>
#include <math.h>

// ---------------------------------------------------------------------------
// Fused 3D local-NCC (window 9) for (2,1,160,160,160) f32 volumes on gfx1250.
// Per 16^3 output tile: stage 24^3 halo tiles of fixed/warped into LDS
// (async-to-LDS), run the 9-tap separable box filter for the five fields as
// banded matmuls on the f32 WMMA pipe (V_WMMA_F32_16X16X4_F32,
// D = A(16x24) * Band(24x16) chained over K), then NCC + clip in registers
// and a deterministic two-stage reduction.
// ---------------------------------------------------------------------------

typedef float v2f __attribute__((ext_vector_type(2)));
typedef float v8f __attribute__((ext_vector_type(8)));

#define RAD   4
#define TOUT  16
#define TIN   24              // TOUT + 2*RAD
#define FXS   25              // padded x-stride of F/W input tiles (coprime w/ 64 banks)
#define NF    (TIN*TIN*FXS)   // 14400 floats per staged input tile
#define S1RS  17              // S1 row stride: addr = (z*24+y)*17 + x'
#define NS1   (TIN*TIN*S1RS)  // 9792
#define S2ZS  272             // 16*17 : S2 addr = z*272 + y'*17 + x'
#define NS2   (TIN*S2ZS)      // 6528
#define SMEM_FLOATS (2*NF + NS1 + NS2)   // 45120 floats = 180480 bytes (<320KB WGP LDS)

#define DIMX 160
#define NTILE 10              // 160/16
#define NBLOCKS (2*NTILE*NTILE*NTILE)
#define EPS 1e-8f

#ifndef __has_builtin
#define __has_builtin(x) 0
#endif
#if __has_builtin(__builtin_amdgcn_global_load_async_to_lds_b32)
#define HAVE_ASYNC_LDS 1
#else
#define HAVE_ASYNC_LDS 0
#endif

#if defined(__HIP_DEVICE_COMPILE__)
// async-to-LDS builtin takes (global int*, lds int*, imm offset, cpol)
typedef __attribute__((address_space(1))) int* as1_i32p;
typedef __attribute__((address_space(3))) int* as3_i32p;
#endif

// Band matrix chunk (4x16 slice of the 24x16 box-filter matrix, scaled 1/9).
// B VGPR layout assumed to mirror the ISA's 32-bit A layout:
//   vgpr g, lane L: row K = 2*(L>>4) + g, col N = L&15.
__device__ __forceinline__ v2f band_chunk(int cc, int lane) {
  const int h = lane >> 4, n = lane & 15;
  const int k0 = 4 * cc + 2 * h;
  v2f b;
  b.x = (k0     >= n && k0     <= n + 8) ? (1.0f / 9.0f) : 0.0f;
  b.y = (k0 + 1 >= n && k0 + 1 <= n + 8) ? (1.0f / 9.0f) : 0.0f;
  return b;
}

__device__ __forceinline__ v8f wmma_f32(v2f a, v2f b, v8f c) {
  // D(16x16,f32) = A(16x4,f32) x B(4x16,f32) + C ; full fp32 precision.
  return __builtin_amdgcn_wmma_f32_16x16x4_f32(false, a, false, b, (short)0, c,
                                               false, false);
}

template <int P>
__device__ __forceinline__ float fval(const float* __restrict__ Ft,
                                      const float* __restrict__ Wt, int off) {
  if (P == 0) return Ft[off];
  if (P == 1) return Wt[off];
  if (P == 2) { const float f = Ft[off]; return f * f; }
  if (P == 3) { const float w = Wt[off]; return w * w; }
  const float f = Ft[off], w = Wt[off];
  return f * w;
}

// One field: 3 separable banded-matmul passes (x,y,z). Results for this
// wave's voxels land in out[2][8] registers (same voxel<->lane mapping for
// every field since the wave/batch assignment is static).
template <int P>
__device__ __forceinline__ void process_field(
    const float* __restrict__ Ft, const float* __restrict__ Wt,
    float* __restrict__ S1, float* __restrict__ S2, const v2f bands[6],
    int wave, int lane, float (&out)[2][8]) {
  const int h = lane >> 4;
  const int ln = lane & 15;

  // ---- pass 1: filter x. rows r=(z*24+y) in [0,576), 36 batches of 16. ----
  for (int rb = wave; rb < 36; rb += 8) {
    const int rowOff = (rb * 16 + ln) * FXS;   // A[m][k] at rowOff + k
    v8f c = {0.f, 0.f, 0.f, 0.f, 0.f, 0.f, 0.f, 0.f};
#pragma unroll
    for (int cc = 0; cc < 6; ++cc) {
      const int k = 4 * cc + 2 * h;
      v2f a;
      a.x = fval<P>(Ft, Wt, rowOff + k);
      a.y = fval<P>(Ft, Wt, rowOff + k + 1);
      c = wmma_f32(a, bands[cc], c);
    }
#pragma unroll
    for (int g = 0; g < 8; ++g) {            // D: M=8h+g (row), N=ln (x')
      const int rr = rb * 16 + 8 * h + g;
      S1[rr * S1RS + ln] = c[g];
    }
  }
  __syncthreads();

  // ---- pass 2: filter y. rows r=(z*16+x'), 24 batches; batch rb -> z=rb. ----
  for (int rb = wave; rb < 24; rb += 8) {
    v8f c = {0.f, 0.f, 0.f, 0.f, 0.f, 0.f, 0.f, 0.f};
#pragma unroll
    for (int cc = 0; cc < 6; ++cc) {
      const int k = 4 * cc + 2 * h;
      v2f a;
      a.x = S1[(rb * 24 + k) * S1RS + ln];     // S1[z=rb][y=k][x'=ln]
      a.y = S1[(rb * 24 + k + 1) * S1RS + ln];
      c = wmma_f32(a, bands[cc], c);
    }
#pragma unroll
    for (int g = 0; g < 8; ++g)              // D: M=x'=8h+g, N=y'=ln
      S2[rb * S2ZS + ln * 17 + 8 * h + g] = c[g];
  }
  __syncthreads();

  // ---- pass 3: filter z. rows r=(y'*16+x'), 16 batches; rb=2*wave+i. ----
#pragma unroll
  for (int i = 0; i < 2; ++i) {
    const int rb = wave * 2 + i;
    v8f c = {0.f, 0.f, 0.f, 0.f, 0.f, 0.f, 0.f, 0.f};
#pragma unroll
    for (int cc = 0; cc < 6; ++cc) {
      const int k = 4 * cc + 2 * h;
      v2f a;
      a.x = S2[k * S2ZS + rb * 17 + ln];       // S2[z=k][y'=rb][x'=ln]
      a.y = S2[(k + 1) * S2ZS + rb * 17 + ln];
      c = wmma_f32(a, bands[cc], c);
    }
#pragma unroll
    for (int g = 0; g < 8; ++g)              // voxel: z'=ln, y'=rb, x'=8h+g
      out[i][g] = c[g];
  }
  __syncthreads();  // S1/S2 safe for reuse by next field
}

__global__ void __launch_bounds__(256)
ncc_tile_kernel(const float* __restrict__ fixedp,
                const float* __restrict__ warpedp,
                float* __restrict__ partial) {
  extern __shared__ float smem[];
  float* Ft = smem;                 // 14400
  float* Wt = smem + NF;            // 14400
  float* S1 = smem + 2 * NF;        // 9792
  float* S2 = smem + 2 * NF + NS1;  // 6528

  const int tid = threadIdx.x;
  const int lane = tid & 31;
  const int wave = tid >> 5;

  int blk = blockIdx.x;
  const int tx = blk % NTILE; blk /= NTILE;
  const int ty = blk % NTILE; blk /= NTILE;
  const int tz = blk % NTILE; blk /= NTILE;
  const int b  = blk;  // 0..1

  const long long vbase = (long long)b * (long long)DIMX * DIMX * DIMX;
  const int gz0 = tz * TOUT - RAD;
  const int gy0 = ty * TOUT - RAD;
  const int gx0 = tx * TOUT - RAD;

  // ---- stage 24^3 halo tiles (zero-padded) into LDS ----
  for (int idx = tid; idx < TIN * TIN * TIN; idx += 256) {
    const int z = idx / (TIN * TIN);
    const int r2 = idx % (TIN * TIN);
    const int y = r2 / TIN, x = r2 % TIN;
    const int gz = gz0 + z, gy = gy0 + y, gx = gx0 + x;
    const int l = (z * TIN + y) * FXS + x;
    if ((unsigned)gz < (unsigned)DIMX && (unsigned)gy < (unsigned)DIMX &&
        (unsigned)gx < (unsigned)DIMX) {
      const long long g = vbase + ((long long)gz * DIMX + gy) * DIMX + gx;
#if HAVE_ASYNC_LDS
      __builtin_amdgcn_global_load_async_to_lds_b32(
          (as1_i32p)(fixedp + g), (as3_i32p)(Ft + l), 0, 0);
      __builtin_amdgcn_global_load_async_to_lds_b32(
          (as1_i32p)(warpedp + g), (as3_i32p)(Wt + l), 0, 0);
#else
      Ft[l] = fixedp[g];
      Wt[l] = warpedp[g];
#endif
    } else {
      Ft[l] = 0.0f;
      Wt[l] = 0.0f;
    }
  }
#if HAVE_ASYNC_LDS
#if __has_builtin(__builtin_amdgcn_s_wait_asynccnt)
  __builtin_amdgcn_s_wait_asynccnt(0);
#else
  asm volatile("s_wait_asynccnt 0" ::: "memory");
#endif
#endif
  __syncthreads();

  // ---- band matrix chunks (constant per lane) ----
  v2f bands[6];
#pragma unroll
  for (int cc = 0; cc < 6; ++cc) bands[cc] = band_chunk(cc, lane);

  // ---- five filtered fields, all in registers per voxel ----
  float a_fm[2][8], a_wm[2][8], a_fs[2][8], a_ws[2][8], a_fw[2][8];
  process_field<0>(Ft, Wt, S1, S2, bands, wave, lane, a_fm);
  process_field<1>(Ft, Wt, S1, S2, bands, wave, lane, a_wm);
  process_field<2>(Ft, Wt, S1, S2, bands, wave, lane, a_fs);
  process_field<3>(Ft, Wt, S1, S2, bands, wave, lane, a_ws);
  process_field<4>(Ft, Wt, S1, S2, bands, wave, lane, a_fw);

  // ---- NCC per voxel, per-thread partial over 16 voxels ----
  float lsum = 0.0f;
#pragma unroll
  for (int i = 0; i < 2; ++i) {
#pragma unroll
    for (int g = 0; g < 8; ++g) {
      const float fm = a_fm[i][g], wm = a_wm[i][g];
      const float fv = a_fs[i][g] - fm * fm;
      const float wv = a_ws[i][g] - wm * wm;
      const float cv = a_fw[i][g] - fm * wm;
      const float den =
          sqrtf(fmaxf(fv, 0.0f) + EPS) * sqrtf(fmaxf(wv, 0.0f) + EPS);
      lsum += fminf(fmaxf(cv / den, -1.0f), 1.0f);
    }
  }

  // ---- block reduction (reuse S1) ----
  __syncthreads();
  S1[tid] = lsum;
  __syncthreads();
#pragma unroll
  for (int s = 128; s > 0; s >>= 1) {
    if (tid < s) S1[tid] += S1[tid + s];
    __syncthreads();
  }
  if (tid == 0) partial[blockIdx.x] = S1[0];
}

__global__ void __launch_bounds__(256)
ncc_reduce_kernel(const float* __restrict__ partial, float* __restrict__ out,
                  int n, float scale) {
  __shared__ float red[256];
  float s = 0.0f;
  for (int i = threadIdx.x; i < n; i += 256) s += partial[i];  // fixed order
  red[threadIdx.x] = s;
  __syncthreads();
#pragma unroll
  for (int st = 128; st > 0; st >>= 1) {
    if (threadIdx.x < st) red[threadIdx.x] += red[threadIdx.x + st];
    __syncthreads();
  }
  if (threadIdx.x == 0) *out = -red[0] * scale;
}

extern "C" void kernel_launch(void* const* d_in, const int* in_sizes, int n_in,
                              void* d_out, int out_size, void* d_ws,
                              size_t ws_size, hipStream_t stream) {
  (void)in_sizes; (void)n_in; (void)out_size; (void)ws_size;
  const float* fixedp  = (const float*)d_in[0];
  const float* warpedp = (const float*)d_in[1];
  float* out = (float*)d_out;
  float* partial = (float*)d_ws;  // NBLOCKS floats of scratch

  const size_t smem_bytes = (size_t)SMEM_FLOATS * sizeof(float);
  hipError_t err = hipFuncSetAttribute(
      (const void*)ncc_tile_kernel,
      hipFuncAttributeMaxDynamicSharedMemorySize, (int)smem_bytes);
  (void)err;

  hipLaunchKernelGGL(ncc_tile_kernel, dim3(NBLOCKS), dim3(256), smem_bytes,
                     stream, fixedp, warpedp, partial);

  const float scale = 1.0f / (2.0f * DIMX * DIMX * DIMX);
  hipLaunchKernelGGL(ncc_reduce_kernel, dim3(1), dim3(256), 0, stream, partial,
                     out, NBLOCKS, scale);
}